// Hadamard_24524263260381
// MI455X (gfx1250) — compile-verified
//
#include <hip/hip_runtime.h>

// FWHT-4096 on MI455X (gfx1250), single pass, fp32, via V_WMMA_F32_16X16X4_F32.
// H4096 = H16 (x) H16 (x) H16 : three digit-passes of D = H16 * B (K=16 as 4xK=4).
// Inbound row copy uses GLOBAL_LOAD_ASYNC_TO_LDS_B128 (ASYNCcnt), outbound uses
// nontemporal stores straight from D fragments. Wave-private LDS -> no barriers.

typedef __attribute__((ext_vector_type(2))) float v2f;
typedef __attribute__((ext_vector_type(8))) float v8f;

#define DIM 4096
#define WAVES_PER_BLOCK 4   // 4 waves * 16KB LDS slice = 64KB static shared

__device__ __forceinline__ float hsign(int m, int k) {
    // H16[m][k] = (-1)^popcount(m & k)
    return (__popc(m & k) & 1) ? -1.0f : 1.0f;
}

__global__ __launch_bounds__(WAVES_PER_BLOCK * 32)
void fwht4096_wmma_kernel(const float* __restrict__ x, float* __restrict__ y)
{
    // Single static LDS variable -> lives at LDS segment offset 0, so raw byte
    // offsets handed to the async-load instruction line up with pointer accesses.
    __shared__ float lds_all[WAVES_PER_BLOCK][DIM];

    const int lane = threadIdx.x & 31;
    const int wave = threadIdx.x >> 5;
    const int row  = blockIdx.x * WAVES_PER_BLOCK + wave;

    const float* __restrict__ src = x + (size_t)row * DIM;
    float* __restrict__ dst       = y + (size_t)row * DIM;
    float* L = lds_all[wave];

    // ---- async global -> LDS copy of the row (no VGPR staging, ASYNCcnt) ----
    {
        const unsigned lds_base = (unsigned)(wave * DIM) * 4u;   // byte offset of slice
        #pragma unroll
        for (int i = 0; i < DIM / (32 * 4); ++i) {
            const int idx       = i * 128 + lane * 4;            // float index in row
            const unsigned loff = lds_base + (unsigned)idx * 4u; // LDS byte offset
            const float* gp     = src + idx;                     // 16B per lane
            asm volatile("global_load_async_to_lds_b128 %0, %1, off th:TH_LOAD_NT"
                         :: "v"(loff), "v"(gp) : "memory");
        }
        asm volatile("s_wait_asynccnt 0x0" ::: "memory");        // LDS slice ready
    }

    // ---- per-lane fragment coordinates (f32 WMMA layouts, wave32) ----
    const int n    = lane & 15;   // N index (B / C / D)
    const int half = lane >> 4;   // 0: lanes 0-15, 1: lanes 16-31
    const int koff = half * 2;    // k_local = koff + vgpr  (A and B share this pattern)
    const int m    = lane & 15;   // M index (A rows)

    // A fragments: H16 split into 4 chunks of K=4 (+/-1 from popcount parity)
    v2f a[4];
    #pragma unroll
    for (int c = 0; c < 4; ++c) {
        a[c].x = hsign(m, 4 * c + koff);
        a[c].y = hsign(m, 4 * c + koff + 1);
    }

    // ---- Stage 1: transform digit n0 (index = t*256 + n1*16 + n0), in-place per tile ----
    #pragma unroll 4
    for (int t = 0; t < 16; ++t) {
        v8f acc = {};
        #pragma unroll
        for (int c = 0; c < 4; ++c) {
            const int k = 4 * c + koff;
            v2f b;
            b.x = L[t * 256 + n * 16 + k];
            b.y = L[t * 256 + n * 16 + k + 1];
            acc = __builtin_amdgcn_wmma_f32_16x16x4_f32(
                false, a[c], false, b, (short)0, acc, false, false);
        }
        #pragma unroll
        for (int v = 0; v < 8; ++v)
            L[t * 256 + n * 16 + (v + 8 * half)] = acc[v];
    }

    // ---- Stage 2: transform digit n1 ----
    #pragma unroll 4
    for (int t = 0; t < 16; ++t) {
        v8f acc = {};
        #pragma unroll
        for (int c = 0; c < 4; ++c) {
            const int k = 4 * c + koff;
            v2f b;
            b.x = L[t * 256 + k * 16 + n];
            b.y = L[t * 256 + (k + 1) * 16 + n];
            acc = __builtin_amdgcn_wmma_f32_16x16x4_f32(
                false, a[c], false, b, (short)0, acc, false, false);
        }
        #pragma unroll
        for (int v = 0; v < 8; ++v)
            L[t * 256 + (v + 8 * half) * 16 + n] = acc[v];
    }

    // ---- Stage 3: transform digit n2; scale by 1/64; nontemporal store to global ----
    #pragma unroll 4
    for (int t = 0; t < 16; ++t) {
        v8f acc = {};
        #pragma unroll
        for (int c = 0; c < 4; ++c) {
            const int k = 4 * c + koff;
            v2f b;
            b.x = L[k * 256 + t * 16 + n];
            b.y = L[(k + 1) * 256 + t * 16 + n];
            acc = __builtin_amdgcn_wmma_f32_16x16x4_f32(
                false, a[c], false, b, (short)0, acc, false, false);
        }
        #pragma unroll
        for (int v = 0; v < 8; ++v)
            __builtin_nontemporal_store(acc[v] * 0.015625f,   // 1/sqrt(4096)
                                        &dst[(v + 8 * half) * 256 + t * 16 + n]);
    }
}

extern "C" void kernel_launch(void* const* d_in, const int* in_sizes, int n_in,
                              void* d_out, int out_size, void* d_ws, size_t ws_size,
                              hipStream_t stream)
{
    (void)n_in; (void)out_size; (void)d_ws; (void)ws_size;
    const float* x = (const float*)d_in[0];
    float* y = (float*)d_out;

    const int rows   = in_sizes[0] / DIM;        // 8 * 2048 = 16384
    const int blocks = rows / WAVES_PER_BLOCK;   // 4096 blocks, 1 row per wave
    fwht4096_wmma_kernel<<<blocks, WAVES_PER_BLOCK * 32, 0, stream>>>(x, y);
}